// Map_88064009437459
// MI455X (gfx1250) — compile-verified
//
#include <hip/hip_runtime.h>

// ---------------------------------------------------------------------------
// Star-map scatter for MI455X (gfx1250).
//
// Memory-bound scatter: AI ~0.3 FLOP/B, so no matrix math applies. Strategy:
//  - zero the 1441x2880 f32 image with B128 stores (16.6 MB, fits in 192MB L2)
//  - stream lat/lng/mags with non-temporal B128 loads (read-once data must
//    not evict the L2-resident output that the random 4B scatters hit)
//  - global_prefetch_b8 (gfx1250 prefetch path) to stay ahead of the stream
//  - 4 stars per thread, 256-thread blocks (8 wave32 waves per block)
// ---------------------------------------------------------------------------

typedef float v4f __attribute__((ext_vector_type(4)));

#define MAP_H 1441
#define MAP_W 2880

__global__ __launch_bounds__(256) void fill_zero_b128(v4f* __restrict__ out, int n4) {
    int i = blockIdx.x * blockDim.x + threadIdx.x;
    if (i < n4) {
        v4f z = {0.0f, 0.0f, 0.0f, 0.0f};
        out[i] = z;   // global_store_b128
    }
}

__global__ __launch_bounds__(256) void fill_zero_tail(float* __restrict__ out,
                                                      int base, int n) {
    int i = base + blockIdx.x * blockDim.x + threadIdx.x;
    if (i < n) out[i] = 0.0f;
}

__global__ __launch_bounds__(256) void star_scatter4(const v4f* __restrict__ lat4,
                                                     const v4f* __restrict__ lng4,
                                                     const v4f* __restrict__ mag4,
                                                     float* __restrict__ out,
                                                     int n4) {
    int i = blockIdx.x * blockDim.x + threadIdx.x;
    if (i >= n4) return;

    // Prefetch ~8KB ahead on each input stream (lowers to global_prefetch_b8).
    if (i + 512 < n4) {
        __builtin_prefetch(&lat4[i + 512], 0, 1);
        __builtin_prefetch(&lng4[i + 512], 0, 1);
        __builtin_prefetch(&mag4[i + 512], 0, 1);
    }

    // Non-temporal streaming loads: read-once data, keep L2 for the output.
    v4f la = __builtin_nontemporal_load(&lat4[i]);
    v4f lo = __builtin_nontemporal_load(&lng4[i]);
    v4f mg = __builtin_nontemporal_load(&mag4[i]);

#pragma unroll
    for (int k = 0; k < 4; ++k) {
        // Same f32 arithmetic as the reference: truncate toward zero (values >= 0).
        int ilat = (int)(la[k] * 8.0f);
        int ilng = (int)(lo[k] * 8.0f);
        if ((unsigned)ilat < (unsigned)MAP_H && (unsigned)ilng < (unsigned)MAP_W) {
            // vertical flip fused into the store address; scale fused into value
            out[(MAP_H - 1 - ilat) * MAP_W + ilng] = mg[k] * 255.0f;
        }
    }
}

__global__ __launch_bounds__(256) void star_scatter_tail(const float* __restrict__ lat,
                                                         const float* __restrict__ lng,
                                                         const float* __restrict__ mag,
                                                         float* __restrict__ out,
                                                         int base, int n) {
    int i = base + blockIdx.x * blockDim.x + threadIdx.x;
    if (i >= n) return;
    int ilat = (int)(lat[i] * 8.0f);
    int ilng = (int)(lng[i] * 8.0f);
    if ((unsigned)ilat < (unsigned)MAP_H && (unsigned)ilng < (unsigned)MAP_W) {
        out[(MAP_H - 1 - ilat) * MAP_W + ilng] = mag[i] * 255.0f;
    }
}

extern "C" void kernel_launch(void* const* d_in, const int* in_sizes, int n_in,
                              void* d_out, int out_size, void* d_ws, size_t ws_size,
                              hipStream_t stream) {
    const float* lat = (const float*)d_in[0];
    const float* lng = (const float*)d_in[1];
    const float* mag = (const float*)d_in[2];
    float* out = (float*)d_out;

    const int n = in_sizes[0];        // 5,000,000 stars

    // --- Phase 1: zero the output image (also clears 0xAA poison) ---------
    const int out4  = out_size >> 2;          // 1,037,520 float4 (divisible)
    const int outr  = out_size & 3;           // 0 here, handled anyway
    if (out4 > 0) {
        fill_zero_b128<<<(out4 + 255) / 256, 256, 0, stream>>>((v4f*)out, out4);
    }
    if (outr > 0) {
        fill_zero_tail<<<1, 256, 0, stream>>>(out, out4 * 4, out_size);
    }

    // --- Phase 2: scatter stars (same stream => ordered after the fill) ---
    const int n4  = n >> 2;                   // 1,250,000 (divisible)
    const int nr  = n & 3;                    // 0 here, handled anyway
    if (n4 > 0) {
        star_scatter4<<<(n4 + 255) / 256, 256, 0, stream>>>(
            (const v4f*)lat, (const v4f*)lng, (const v4f*)mag, out, n4);
    }
    if (nr > 0) {
        star_scatter_tail<<<1, 256, 0, stream>>>(lat, lng, mag, out, n4 * 4, n);
    }
}